// RolloutStorage_67181878444991
// MI455X (gfx1250) — compile-verified
//
#include <hip/hip_runtime.h>
#include <hip/hip_bf16.h>

#define T_DIM   128
#define BLK1    128     // K1 block (4 wave32s; 512 workgroups -> good WGP spread)
#define BLK     256     // K2/K3 block
#define STAGES  8       // async->LDS pipeline depth (12KB staging LDS per block)

#ifndef __AS
#define __AS(n) __attribute__((address_space(n)))
#endif

// ---- CDNA5 async global->LDS copy (one b32 per lane), ASYNCcnt-tracked ----
__device__ __forceinline__ void async_b32(const void* g, void* l) {
#if __has_builtin(__builtin_amdgcn_global_load_async_to_lds_b32)
    void* gnc = const_cast<void*>(g);
    __builtin_amdgcn_global_load_async_to_lds_b32(
        (__AS(1) int*)gnc, (__AS(3) int*)l, 0, 0);
#else
    unsigned lds_addr = (unsigned)(unsigned long long)l;
    asm volatile("global_load_async_to_lds_b32 %0, %1, off"
                 :
                 : "v"(lds_addr), "v"(g)
                 : "memory");
#endif
}

// asm waits (not builtins) so the "memory" clobber orders surrounding LDS ops
#define WAIT_ASYNC(n) asm volatile("s_wait_asynccnt %0" :: "i"(n) : "memory")
#define WAIT_DS0()    asm volatile("s_wait_dscnt 0" ::: "memory")

// ---------------------------------------------------------------------------
// K1: backward GAE scan. One thread per env. 8-stage async->LDS pipeline.
// Writes returns and raw advantages; emits per-block (sum, sumsq) doubles.
// ---------------------------------------------------------------------------
__global__ void __launch_bounds__(BLK1) gae_scan_kernel(
    const float* __restrict__ rewards,
    const float* __restrict__ values,
    const float* __restrict__ last_values,
    const int*   __restrict__ dones,
    float* __restrict__ ret_out,
    float* __restrict__ adv_out,
    double* __restrict__ partials,
    int N)
{
    __shared__ float  sh_r[STAGES][BLK1];
    __shared__ float  sh_v[STAGES][BLK1];
    __shared__ int    sh_d[STAGES][BLK1];
    __shared__ double red_s[BLK1];
    __shared__ double red_q[BLK1];

    const int tid = threadIdx.x;
    const int n   = blockIdx.x * BLK1 + tid;

    // Prologue: launch async stages for t = T-1 .. T-STAGES
#pragma unroll
    for (int s = 0; s < STAGES; ++s) {
        const int t = T_DIM - 1 - s;
        const size_t off = (size_t)t * (size_t)N + (size_t)n;
        async_b32(rewards + off, &sh_r[s][tid]);
        async_b32(values  + off, &sh_v[s][tid]);
        async_b32(dones   + off, &sh_d[s][tid]);
    }

    float v_next = last_values[n];
    float adv    = 0.0f;
    float acc_s  = 0.0f;
    float acc_q  = 0.0f;
    const float gamma = 0.99f;
    const float glam  = (float)(0.99 * 0.95);

    // Steady state: t = T-1 .. STAGES. Oldest stage complete when
    // asynccnt <= 3*(STAGES-1); always refill the consumed slot with t-STAGES.
#pragma unroll 8
    for (int t = T_DIM - 1; t >= STAGES; --t) {
        const int slot = (T_DIM - 1 - t) & (STAGES - 1);
        WAIT_ASYNC(3 * (STAGES - 1));

        const float r = sh_r[slot][tid];
        const float v = sh_v[slot][tid];
        const int   d = sh_d[slot][tid];
        WAIT_DS0();  // LDS reads landed -> safe to overwrite this slot

        const size_t noff = (size_t)(t - STAGES) * (size_t)N + (size_t)n;
        async_b32(rewards + noff, &sh_r[slot][tid]);
        async_b32(values  + noff, &sh_v[slot][tid]);
        async_b32(dones   + noff, &sh_d[slot][tid]);

        const float nt    = d ? 0.0f : 1.0f;
        const float delta = r + nt * gamma * v_next - v;
        adv = delta + nt * glam * adv;

        const size_t off = (size_t)t * (size_t)N + (size_t)n;
        ret_out[off] = adv + v;
        adv_out[off] = adv;

        acc_s += adv;
        acc_q += adv * adv;
        v_next = v;
    }

    // Tail: all stages already issued; drain everything once, then consume.
#pragma unroll
    for (int t = STAGES - 1; t >= 0; --t) {
        const int slot = (T_DIM - 1 - t) & (STAGES - 1);
        WAIT_ASYNC(0);

        const float r = sh_r[slot][tid];
        const float v = sh_v[slot][tid];
        const int   d = sh_d[slot][tid];

        const float nt    = d ? 0.0f : 1.0f;
        const float delta = r + nt * gamma * v_next - v;
        adv = delta + nt * glam * adv;

        const size_t off = (size_t)t * (size_t)N + (size_t)n;
        ret_out[off] = adv + v;
        adv_out[off] = adv;

        acc_s += adv;
        acc_q += adv * adv;
        v_next = v;
    }

    // Deterministic fixed-order block reduction (double precision partials)
    red_s[tid] = (double)acc_s;
    red_q[tid] = (double)acc_q;
    __syncthreads();
#pragma unroll
    for (int w = BLK1 / 2; w > 0; w >>= 1) {
        if (tid < w) {
            red_s[tid] += red_s[tid + w];
            red_q[tid] += red_q[tid + w];
        }
        __syncthreads();
    }
    if (tid == 0) {
        partials[2 * blockIdx.x]     = red_s[0];
        partials[2 * blockIdx.x + 1] = red_q[0];
    }
}

// ---------------------------------------------------------------------------
// K2: single-block deterministic reduction of partials -> mean, 1/(std+1e-8)
// ---------------------------------------------------------------------------
__global__ void __launch_bounds__(BLK) gae_stats_kernel(
    const double* __restrict__ partials, int nblocks,
    float* __restrict__ stats, long long count)
{
    __shared__ double ss[BLK];
    __shared__ double qq[BLK];
    const int i = threadIdx.x;
    double s = 0.0, q = 0.0;
    for (int b = i; b < nblocks; b += BLK) {
        s += partials[2 * b];
        q += partials[2 * b + 1];
    }
    ss[i] = s; qq[i] = q;
    __syncthreads();
#pragma unroll
    for (int w = BLK / 2; w > 0; w >>= 1) {
        if (i < w) { ss[i] += ss[i + w]; qq[i] += qq[i + w]; }
        __syncthreads();
    }
    if (i == 0) {
        const double cnt  = (double)count;
        const double mean = ss[0] / cnt;
        double var = (qq[0] - ss[0] * ss[0] / cnt) / (cnt - 1.0);  // ddof=1
        if (var < 0.0) var = 0.0;
        const double stdv = sqrt(var);
        stats[0] = (float)mean;
        stats[1] = (float)(1.0 / (stdv + 1e-8));
    }
}

// ---------------------------------------------------------------------------
// K3: normalize advantages in place (float4 vectorized)
// ---------------------------------------------------------------------------
__global__ void __launch_bounds__(BLK) gae_norm_kernel(
    float* __restrict__ adv, const float* __restrict__ stats, int n4)
{
    const int i = blockIdx.x * BLK + threadIdx.x;
    if (i >= n4) return;
    const float mean  = stats[0];
    const float scale = stats[1];
    float4* p = (float4*)adv;
    float4 a = p[i];
    a.x = (a.x - mean) * scale;
    a.y = (a.y - mean) * scale;
    a.z = (a.z - mean) * scale;
    a.w = (a.w - mean) * scale;
    p[i] = a;
}

// ---------------------------------------------------------------------------
extern "C" void kernel_launch(void* const* d_in, const int* in_sizes, int n_in,
                              void* d_out, int out_size, void* d_ws, size_t ws_size,
                              hipStream_t stream) {
    const float* rewards = (const float*)d_in[0];  // [T,N,1]
    const float* values  = (const float*)d_in[1];  // [T,N,1]
    const float* last_v  = (const float*)d_in[2];  // [N,1]
    const int*   dones   = (const int*)d_in[3];    // [T,N,1]

    const int N  = in_sizes[2];   // 65536
    const int TN = in_sizes[0];   // T*N

    float* ret_out = (float*)d_out;        // first half: returns
    float* adv_out = ret_out + TN;         // second half: (raw then normalized) adv

    double* partials = (double*)d_ws;
    const int nblocks = N / BLK1;          // 512
    float* stats = (float*)((char*)d_ws + (size_t)nblocks * 2 * sizeof(double));

    gae_scan_kernel<<<nblocks, BLK1, 0, stream>>>(
        rewards, values, last_v, dones, ret_out, adv_out, partials, N);

    gae_stats_kernel<<<1, BLK, 0, stream>>>(partials, nblocks, stats, (long long)TN);

    const int n4 = TN / 4;
    gae_norm_kernel<<<(n4 + BLK - 1) / BLK, BLK, 0, stream>>>(adv_out, stats, n4);
}